// Repro_10806137717294
// MI455X (gfx1250) — compile-verified
//
#include <hip/hip_runtime.h>
#include <hip/hip_bf16.h>

// y = 2*x for N=100 f32 elements, computed as a WMMA matmul against a scaled
// identity: Y(16x16) = sum_k X[:,4k:4k+4] * (2*I)[4k:4k+4,:]  (bit-exact in f32).
// One wave32 per 256-element tile; emits 4x v_wmma_f32_16x16x4_f32.

typedef __attribute__((ext_vector_type(2))) float v2f;
typedef __attribute__((ext_vector_type(8))) float v8f;

__global__ __launch_bounds__(32) void
scale2_wmma_kernel(const float* __restrict__ x, float* __restrict__ y, int n) {
    const int lane = threadIdx.x;        // 0..31, wave32
    const int base = blockIdx.x * 256;   // 16x16 tile per block
    const int m    = lane & 15;          // A: row M (lanes 0-15 and 16-31 both map M=lane%16)
                                         // B/D: column N = lane%16
    const int hiK  = (lane >= 16) ? 2 : 0;
    const int hiM  = (lane >= 16) ? 8 : 0;

    v8f c = {};  // zero-initialized f32 accumulator (C matrix)

#pragma unroll
    for (int k = 0; k < 4; ++k) {
        v2f a, b;
#pragma unroll
        for (int v = 0; v < 2; ++v) {
            // A fragment: A[m][kcol] = X[m][4k + kcol], X row-major over the tile
            const int kcol = v + hiK;              // 0..3 within this K-slice
            const int idx  = base + m * 16 + (4 * k + kcol);
            const float val = x[(idx < n) ? idx : 0];   // clamped (always in-bounds) load
            a[v] = (idx < n) ? val : 0.0f;              // zero-pad tail

            // B fragment: (2*I) slice, rows 4k..4k+3. Row krow, col n=m.
            const int krow = v + hiK;
            b[v] = (m == (4 * k + krow)) ? 2.0f : 0.0f;
        }
        // 8 args: (neg_a, A, neg_b, B, c_mod, C, reuse_a, reuse_b)
        c = __builtin_amdgcn_wmma_f32_16x16x4_f32(
                /*neg_a=*/false, a, /*neg_b=*/false, b,
                /*c_mod=*/(short)0, c, /*reuse_a=*/false, /*reuse_b=*/false);
    }

    // D layout: VGPR r -> M = r + (lane>=16 ? 8 : 0), N = lane%16
#pragma unroll
    for (int r = 0; r < 8; ++r) {
        const int idx = base + (r + hiM) * 16 + m;
        if (idx < n) y[idx] = c[r];
    }
}

extern "C" void kernel_launch(void* const* d_in, const int* in_sizes, int n_in,
                              void* d_out, int out_size, void* d_ws, size_t ws_size,
                              hipStream_t stream) {
    const float* x = (const float*)d_in[0];
    float* y       = (float*)d_out;
    const int n    = in_sizes[0];            // 100

    const int blocks = (n + 255) / 256;      // 1 block for n=100
    scale2_wmma_kernel<<<blocks, 32, 0, stream>>>(x, y, n);
}